// MapMemorizer_13116830122725
// MI455X (gfx1250) — compile-verified
//
#include <hip/hip_runtime.h>
#include <math.h>

typedef float v2f __attribute__((ext_vector_type(2)));
typedef float v8f __attribute__((ext_vector_type(8)));
typedef int   v4i __attribute__((__vector_size__(4 * sizeof(int))));

#define WMMA_F32(a, b, c) \
  __builtin_amdgcn_wmma_f32_16x16x4_f32(false, (a), false, (b), (short)0, (c), false, false)

// ---- optional CDNA5 async global->LDS path -----------------------------
#if defined(__has_builtin)
#if __has_builtin(__builtin_amdgcn_global_load_async_to_lds_b128) && \
    __has_builtin(__builtin_amdgcn_s_wait_asynccnt)
#define HAVE_ASYNC_LDS 1
#endif
#endif

#ifdef HAVE_ASYNC_LDS
// Probe-confirmed: param 1 is a generic pointer to a 16-byte int vector.
__device__ __forceinline__ void async_copy16(const float* g, float* l) {
  __builtin_amdgcn_global_load_async_to_lds_b128((v4i*)g, (v4i*)l, 0, 0);
}
__device__ __forceinline__ void async_wait_all() {
  __builtin_amdgcn_s_wait_asynccnt(0);
}
#endif

// ---- problem constants -------------------------------------------------
constexpr int BB   = 16;           // batch
constexpr int LL   = 4096;         // sequence
constexpr int DM   = 64;           // d_model
constexpr int DI   = 128;          // d_inner
constexpr int DSTATE = 32;         // d_state
constexpr int NH   = 2;            // heads
constexpr int HD   = 64;           // headdim
constexpr int CONV = DI + 2 * DSTATE;          // 192
constexpr int DPJ  = 2 * DI + 2 * DSTATE + NH; // 322
constexpr int Q    = 64;           // chunk length
constexpr int NCH  = LL / Q;       // 64 chunks
constexpr int BH   = BB * NH;      // 32
constexpr int ROWS = BB * LL;      // 65536

__device__ __forceinline__ float siluf(float x) { return x / (1.0f + __expf(-x)); }
__device__ __forceinline__ float softplusf(float x) {
  return (x > 20.0f) ? x : log1pf(__expf(x));
}

// ---- LDS-staged fp32 GEMM via V_WMMA_F32_16X16X4_F32 -------------------
// C[M,N] = A[M,K] @ W[K,N]; M % 64 == 0, K in {64,128}, N arbitrary.
// Block: 256 threads = 8 waves computing a 64x64 tile (16 sub-tiles).
__global__ __launch_bounds__(256) void gemm_f32_wmma(
    const float* __restrict__ A, const float* __restrict__ W,
    float* __restrict__ C, int M, int N, int K) {
  __shared__ float Al[64][132];   // [m][k], padded: stride 132 spreads banks
  __shared__ float Wl[128][64];   // [k][n], row-wise reads -> conflict-free
  const int tid = threadIdx.x;
  const int m_base = blockIdx.y * 64;
  const int n_base = blockIdx.x * 64;

  // stage A (64 x K): 16B vector loads, fully coalesced, always in-bounds
  const int nA4 = 64 * K / 4;
  for (int i = tid; i < nA4; i += 256) {
    int i4 = i * 4;
    int r = i4 / K, kc = i4 % K;
    const float* gp = &A[(m_base + r) * K + kc];
#ifdef HAVE_ASYNC_LDS
    async_copy16(gp, &Al[r][kc]);
#else
    *(float4*)(&Al[r][kc]) = *(const float4*)gp;
#endif
  }
  // stage W (K x 64): ragged N handled once here, branch-free k-loop below
  for (int i = tid; i < K * 64; i += 256) {
    int k = i >> 6, c = i & 63;
    int n = n_base + c;
    Wl[k][c] = (n < N) ? W[k * N + n] : 0.0f;
  }
#ifdef HAVE_ASYNC_LDS
  async_wait_all();
#endif
  __syncthreads();

  const int w = tid >> 5, lane = tid & 31;
  const int half = lane >> 4, lm = lane & 15;
  for (int tt = w; tt < 16; tt += 8) {
    const int m0 = (tt >> 2) * 16;
    const int nc0 = (tt & 3) * 16;
    v8f acc = {};
    for (int kk = 0; kk < K; kk += 4) {
      int ka = kk + 2 * half;
      v2f a, bf;
      a.x = Al[m0 + lm][ka];   a.y = Al[m0 + lm][ka + 1];
      bf.x = Wl[ka][nc0 + lm]; bf.y = Wl[ka + 1][nc0 + lm];
      acc = WMMA_F32(a, bf, acc);
    }
    int gn = n_base + nc0 + lm;
    if (gn < N) {
      for (int r = 0; r < 8; ++r)
        C[(m_base + m0 + r + 8 * half) * N + gn] = acc[r];
    }
  }
}

// ---- depthwise causal conv(4) + SiLU, dt = softplus, logdA -------------
__global__ void conv_dt_kernel(
    const float* __restrict__ zx, const float* __restrict__ conv_w,
    const float* __restrict__ conv_b, const float* __restrict__ dt_bias,
    const float* __restrict__ A_log, float* __restrict__ xbc,
    float* __restrict__ dtb, float* __restrict__ lda) {
  const int CPT = CONV + NH; // 194 lanes of work per (b,l)
  int gid = blockIdx.x * blockDim.x + threadIdx.x;
  if (gid >= ROWS * CPT) return;
  int c = gid % CPT;
  int row = gid / CPT;      // row = b*LL + l
  int l = row % LL, b = row / LL;
  if (c < CONV) {
    float s = conv_b[c];
    #pragma unroll
    for (int j = 0; j < 4; ++j) {
      int lj = l - 3 + j;
      float xv = (lj >= 0) ? zx[(b * LL + lj) * DPJ + DI + c] : 0.0f;
      s += xv * conv_w[c * 4 + j];
    }
    xbc[row * CONV + c] = siluf(s);
  } else {
    int h = c - CONV;
    float raw = zx[row * DPJ + (DI + CONV) + h] + dt_bias[h];
    float dt = softplusf(raw);
    float Aneg = -__expf(A_log[h]);
    int bh = b * NH + h;
    dtb[bh * LL + l] = dt;
    lda[bh * LL + l] = dt * Aneg; // log(dA)
  }
}

// ---- per-chunk inclusive cumsum of log(dA) -----------------------------
__global__ void chunk_cumsum_kernel(const float* __restrict__ lda,
                                    float* __restrict__ clog,
                                    float* __restrict__ ctot) {
  int gid = blockIdx.x * blockDim.x + threadIdx.x; // [0, BH*NCH)
  if (gid >= BH * NCH) return;
  int ch = gid % NCH, bh = gid / NCH;
  const float* p = lda + bh * LL + ch * Q;
  float* q = clog + bh * LL + ch * Q;
  float acc = 0.0f;
  for (int t = 0; t < Q; ++t) { acc += p[t]; q[t] = acc; }
  ctot[gid] = acc;
}

// ---- chunk state summary: S_c[n,p] = sum_t exp(tot-clog_t)*dt_t*B_t[n]*x_t[p]
__global__ __launch_bounds__(256) void chunk_state_kernel(
    const float* __restrict__ xbc, const float* __restrict__ dtb,
    const float* __restrict__ clog, const float* __restrict__ ctot,
    float* __restrict__ Schunk) {
  __shared__ float Bw[DSTATE][Q + 4]; // A-matrix layout [n][t], padded
  __shared__ float Xs[Q][HD];         // [t][p]
  __shared__ float wv[Q];
  const int bhch = blockIdx.x;
  const int ch = bhch % NCH, bh = bhch / NCH;
  const int b = bh / NH, h = bh % NH;
  const int tid = threadIdx.x;
  if (tid < Q) {
    float tot = ctot[bhch];
    wv[tid] = __expf(tot - clog[bh * LL + ch * Q + tid]) * dtb[bh * LL + ch * Q + tid];
  }
  __syncthreads();
  for (int i = tid; i < DSTATE * Q; i += 256) {
    int n = i / Q, t = i % Q;
    Bw[n][t] = xbc[(b * LL + ch * Q + t) * CONV + DI + n] * wv[t];
  }
  for (int i = tid; i < Q * HD; i += 256) {
    int t = i >> 6, p = i & 63;
    Xs[t][p] = xbc[(b * LL + ch * Q + t) * CONV + h * HD + p];
  }
  __syncthreads();
  const int w = tid >> 5, lane = tid & 31;
  const int half = lane >> 4, lm = lane & 15;
  const int m0 = (w >> 2) * 16, p0 = (w & 3) * 16; // 2x4 tiles, 8 waves
  v8f acc = {};
  for (int kk = 0; kk < Q; kk += 4) {
    int ka = kk + 2 * half;
    v2f a, bf;
    a.x = Bw[m0 + lm][ka];  a.y = Bw[m0 + lm][ka + 1];
    bf.x = Xs[ka][p0 + lm]; bf.y = Xs[ka + 1][p0 + lm];
    acc = WMMA_F32(a, bf, acc);
  }
  float* S = Schunk + (size_t)bhch * DSTATE * HD;
  for (int r = 0; r < 8; ++r)
    S[(m0 + r + 8 * half) * HD + p0 + lm] = acc[r];
}

// ---- inter-chunk recurrence (64 sequential steps over 32x64 state) -----
__global__ __launch_bounds__(512) void state_scan_kernel(
    const float* __restrict__ Schunk, const float* __restrict__ ctot,
    float* __restrict__ Sprev) {
  const int bh = blockIdx.x;      // 32 blocks
  const int e = threadIdx.x * 4;  // 512 threads * 4 = 2048 elems
  float s0 = 0.f, s1 = 0.f, s2 = 0.f, s3 = 0.f;
  for (int ch = 0; ch < NCH; ++ch) {
    size_t base = (size_t)(bh * NCH + ch) * DSTATE * HD + e;
    if (ch + 1 < NCH)
      __builtin_prefetch(&Schunk[base + DSTATE * HD], 0, 1); // global_prefetch
    Sprev[base + 0] = s0; Sprev[base + 1] = s1;
    Sprev[base + 2] = s2; Sprev[base + 3] = s3;
    float tot = __expf(ctot[bh * NCH + ch]);
    s0 = tot * s0 + Schunk[base + 0];
    s1 = tot * s1 + Schunk[base + 1];
    s2 = tot * s2 + Schunk[base + 2];
    s3 = tot * s3 + Schunk[base + 3];
  }
}

// ---- chunk output: Y = (mask*decay*(C@B^T))@X + (C*cum)@Sprev + Dp*X ---
__global__ __launch_bounds__(256) void chunk_output_kernel(
    const float* __restrict__ xbc, const float* __restrict__ dtb,
    const float* __restrict__ clog, const float* __restrict__ Sprev,
    const float* __restrict__ Dp, float* __restrict__ ybuf) {
  __shared__ float Ct[Q][DSTATE + 1];  // padded against column-read conflicts
  __shared__ float Bt[Q][DSTATE + 1];
  __shared__ float Xs[Q][HD];
  __shared__ float Wl[Q][Q + 4];
  __shared__ float Sp[DSTATE][HD];
  __shared__ float clg[Q], dts[Q], cume[Q];
  const int bhch = blockIdx.x;
  const int ch = bhch % NCH, bh = bhch / NCH;
  const int b = bh / NH, h = bh % NH;
  const int tid = threadIdx.x;
  if (tid < Q) {
    float cv = clog[bh * LL + ch * Q + tid];
    clg[tid] = cv;
    cume[tid] = __expf(cv);
    dts[tid] = dtb[bh * LL + ch * Q + tid];
  }
  for (int i = tid; i < Q * DSTATE; i += 256) {
    int t = i >> 5, n = i & 31;
    int base = (b * LL + ch * Q + t) * CONV;
    Bt[t][n] = xbc[base + DI + n];
    Ct[t][n] = xbc[base + DI + DSTATE + n];
  }
  for (int i = tid; i < Q * HD; i += 256) {
    int t = i >> 6, p = i & 63;
    Xs[t][p] = xbc[(b * LL + ch * Q + t) * CONV + h * HD + p];
  }
  for (int i = tid; i < DSTATE * HD; i += 256)
    Sp[i >> 6][i & 63] = Sprev[(size_t)bhch * DSTATE * HD + i];
  __syncthreads();

  const int w = tid >> 5, lane = tid & 31;
  const int half = lane >> 4, lm = lane & 15;

  // GEMM1: G = Ct(64x32) @ Bt^T(32x64), then decay-mask into Wl
  for (int tt = w; tt < 16; tt += 8) {
    int m0 = (tt >> 2) * 16, s0 = (tt & 3) * 16;
    v8f g = {};
    for (int kk = 0; kk < DSTATE; kk += 4) {
      int ka = kk + 2 * half;
      v2f a, bf;
      a.x = Ct[m0 + lm][ka];  a.y = Ct[m0 + lm][ka + 1];
      bf.x = Bt[s0 + lm][ka]; bf.y = Bt[s0 + lm][ka + 1]; // B^T[k][s] = Bt[s][k]
      g = WMMA_F32(a, bf, g);
    }
    for (int r = 0; r < 8; ++r) {
      int t = m0 + r + 8 * half, s = s0 + lm;
      Wl[t][s] = (s <= t) ? __expf(clg[t] - clg[s]) * dts[s] * g[r] : 0.0f;
    }
  }
  __syncthreads();

  // GEMM2+3 fused: Y = Wl@Xs + (Ct*cume)@Sp ; then + Dp*Xs
  const float dph = Dp[h];
  for (int tt = w; tt < 16; tt += 8) {
    int m0 = (tt >> 2) * 16, p0 = (tt & 3) * 16;
    v8f acc = {};
    for (int kk = 0; kk < Q; kk += 4) {
      int ka = kk + 2 * half;
      v2f a, bf;
      a.x = Wl[m0 + lm][ka];  a.y = Wl[m0 + lm][ka + 1];
      bf.x = Xs[ka][p0 + lm]; bf.y = Xs[ka + 1][p0 + lm];
      acc = WMMA_F32(a, bf, acc);
    }
    float cm = cume[m0 + lm];
    for (int kk = 0; kk < DSTATE; kk += 4) {
      int ka = kk + 2 * half;
      v2f a, bf;
      a.x = Ct[m0 + lm][ka] * cm;  a.y = Ct[m0 + lm][ka + 1] * cm;
      bf.x = Sp[ka][p0 + lm];      bf.y = Sp[ka + 1][p0 + lm];
      acc = WMMA_F32(a, bf, acc);
    }
    for (int r = 0; r < 8; ++r) {
      int t = m0 + r + 8 * half, p = p0 + lm;
      ybuf[(b * LL + ch * Q + t) * DI + h * HD + p] = acc[r] + dph * Xs[t][p];
    }
  }
}

// ---- gate (silu(z)) + RMSNorm ------------------------------------------
__global__ __launch_bounds__(128) void gate_norm_kernel(
    const float* __restrict__ ybuf, const float* __restrict__ zx,
    const float* __restrict__ norm_w, float* __restrict__ ynorm) {
  const int row = blockIdx.x; // b*LL + l
  const int d = threadIdx.x;  // 128 threads = 4 waves
  __shared__ float red[4];
  float z = zx[row * DPJ + d];
  float y = ybuf[row * DI + d] * siluf(z);
  float ss = y * y;
  for (int off = 16; off > 0; off >>= 1) ss += __shfl_down(ss, off, 32);
  if ((d & 31) == 0) red[d >> 5] = ss;
  __syncthreads();
  float tot = red[0] + red[1] + red[2] + red[3];
  ynorm[row * DI + d] = y * rsqrtf(tot * (1.0f / DI) + 1e-5f) * norm_w[d];
}

// ---- host-side orchestration -------------------------------------------
extern "C" void kernel_launch(void* const* d_in, const int* in_sizes, int n_in,
                              void* d_out, int out_size, void* d_ws, size_t ws_size,
                              hipStream_t stream) {
  const float* x       = (const float*)d_in[0];
  const float* Win     = (const float*)d_in[1];
  const float* conv_w  = (const float*)d_in[2];
  const float* conv_b  = (const float*)d_in[3];
  const float* dt_bias = (const float*)d_in[4];
  const float* A_log   = (const float*)d_in[5];
  const float* Dp      = (const float*)d_in[6];
  const float* norm_w  = (const float*)d_in[7];
  const float* Wout    = (const float*)d_in[8];

  float* ws = (float*)d_ws;
  size_t off = 0;
  float* hA     = ws + off; off += (size_t)ROWS * DM;
  float* hB     = ws + off; off += (size_t)ROWS * DM;
  float* zx     = ws + off; off += (size_t)ROWS * DPJ;
  float* xbc    = ws + off; off += (size_t)ROWS * CONV;
  float* dtb    = ws + off; off += (size_t)BH * LL;
  float* lda    = ws + off; off += (size_t)BH * LL;
  float* clog   = ws + off; off += (size_t)BH * LL;
  float* ctot   = ws + off; off += 4096;
  float* Schunk = ws + off; off += (size_t)BH * NCH * DSTATE * HD;
  float* Sprev  = ws + off; off += (size_t)BH * NCH * DSTATE * HD;
  float* ybuf   = ws + off; off += (size_t)ROWS * DI;
  float* ynorm  = ws + off; off += (size_t)ROWS * DI;
  (void)ws_size; (void)in_sizes; (void)n_in; (void)out_size;

  for (int layer = 0; layer < 8; ++layer) {
    const float* in = (layer == 0) ? x : ((layer & 1) ? hA : hB);
    float* out = (layer == 7) ? (float*)d_out : ((layer & 1) ? hB : hA);

    // 1) in-projection: zx = in(65536x64) @ Win(64x322)
    {
      dim3 grid((DPJ + 63) / 64, ROWS / 64);
      gemm_f32_wmma<<<grid, 256, 0, stream>>>(in, Win + (size_t)layer * DM * DPJ,
                                              zx, ROWS, DPJ, DM);
    }
    // 2) conv + silu + dt/softplus + log(dA)
    {
      int total = ROWS * (CONV + NH);
      conv_dt_kernel<<<(total + 255) / 256, 256, 0, stream>>>(
          zx, conv_w + (size_t)layer * CONV * 4, conv_b + (size_t)layer * CONV,
          dt_bias + layer * NH, A_log + layer * NH, xbc, dtb, lda);
    }
    // 3) per-chunk cumsum of log(dA)
    chunk_cumsum_kernel<<<(BH * NCH + 255) / 256, 256, 0, stream>>>(lda, clog, ctot);
    // 4) chunk state summaries (WMMA)
    chunk_state_kernel<<<BH * NCH, 256, 0, stream>>>(xbc, dtb, clog, ctot, Schunk);
    // 5) 64-step inter-chunk recurrence
    state_scan_kernel<<<BH, 512, 0, stream>>>(Schunk, ctot, Sprev);
    // 6) chunk outputs (3 fused WMMA GEMMs + decay mask + skip)
    chunk_output_kernel<<<BH * NCH, 256, 0, stream>>>(xbc, dtb, clog, Sprev,
                                                      Dp + layer * NH, ybuf);
    // 7) gate + RMSNorm
    gate_norm_kernel<<<ROWS, 128, 0, stream>>>(ybuf, zx, norm_w + (size_t)layer * DI, ynorm);
    // 8) out-projection: out = ynorm(65536x128) @ Wout(128x64)
    {
      dim3 grid(1, ROWS / 64);
      gemm_f32_wmma<<<grid, 256, 0, stream>>>(ynorm, Wout + (size_t)layer * DI * DM,
                                              out, ROWS, DM, DI);
    }
  }
}